// HdcLevelEncoder_24472723652908
// MI455X (gfx1250) — compile-verified
//
#include <hip/hip_runtime.h>
#include <stdint.h>

#define N_SAMP   32768
#define DIM      2048
#define LEVELS   100
#define TSTAMP   32768
#define P_BLOCKS 256
#define SPB      (N_SAMP / P_BLOCKS)   // 128 samples per accumulate block

typedef __attribute__((ext_vector_type(2))) float v2f;
typedef __attribute__((ext_vector_type(4))) float v4f;
typedef __attribute__((ext_vector_type(8))) float v8f;

// ---------------------------------------------------------------------------
// Level quantization: round-to-nearest-even (matches jnp.round), clamp.
// ---------------------------------------------------------------------------
__device__ __forceinline__ int level_idx(float v, float lo, float hi, int n) {
    float f = (v - lo) / (hi - lo) * (float)(n - 1);
    int i = (int)rintf(f);
    i = i < 0 ? 0 : i;
    i = i > n - 1 ? n - 1 : i;
    return i;
}

__device__ __forceinline__ float clamp33(float v) {
    return fminf(fmaxf(v, -3.0f), 3.0f);
}

// ---------------------------------------------------------------------------
// Pass 1: per-sample index computation.  idx layout: [9][N_SAMP]
//   0:x 1:y 2:z 3:mag 4:xj 5:yj 6:zj 7:jmag 8:t
// ---------------------------------------------------------------------------
__global__ __launch_bounds__(256) void idx_kernel(const float* __restrict__ inp,
                                                  int* __restrict__ idx) {
    int n = blockIdx.x * blockDim.x + threadIdx.x;
    if (n >= N_SAMP) return;
    float t0 = inp[0];
    float4 cur = ((const float4*)inp)[n];
    float t = cur.x - t0;

    float x = clamp33(cur.y);
    float y = clamp33(cur.z);
    float z = clamp33(cur.w);
    float mag = sqrtf(cur.y * cur.y + cur.z * cur.z + cur.w * cur.w);

    float jx = 0.0f, jy = 0.0f, jz = 0.0f;
    if (n > 0) {
        float4 prev = ((const float4*)inp)[n - 1];
        float dt = cur.x - prev.x;           // == t[n]-t[n-1]
        jx = (cur.y - prev.y) / dt;
        jy = (cur.z - prev.z) / dt;
        jz = (cur.w - prev.w) / dt;
    }
    float jmag = sqrtf(jx * jx + jy * jy + jz * jz);
    float xj = clamp33(jx), yj = clamp33(jy), zj = clamp33(jz);

    idx[0 * N_SAMP + n] = level_idx(x,    -3.0f, 3.0f, LEVELS);
    idx[1 * N_SAMP + n] = level_idx(y,    -3.0f, 3.0f, LEVELS);
    idx[2 * N_SAMP + n] = level_idx(z,    -3.0f, 3.0f, LEVELS);
    idx[3 * N_SAMP + n] = level_idx(mag,  -3.0f, 3.0f, LEVELS);
    idx[4 * N_SAMP + n] = level_idx(xj,   -3.0f, 3.0f, LEVELS);
    idx[5 * N_SAMP + n] = level_idx(yj,   -3.0f, 3.0f, LEVELS);
    idx[6 * N_SAMP + n] = level_idx(zj,   -3.0f, 3.0f, LEVELS);
    idx[7 * N_SAMP + n] = level_idx(jmag, -3.0f, 3.0f, LEVELS);
    idx[8 * N_SAMP + n] = level_idx(t,     0.0f, (float)TSTAMP, TSTAMP);
}

// ---------------------------------------------------------------------------
// Async DMA of one 8KB W_t row into LDS: each of 256 lanes issues two
// global_load_async_to_lds_b128 covering bytes [tid*16,+16) and [4K+tid*16,+16).
// Each lane later reads exactly the bytes it issued, so a per-wave
// s_wait_asynccnt suffices (no workgroup barrier).
// ---------------------------------------------------------------------------
__device__ __forceinline__ void prefetch_row(const float* __restrict__ Wt, int row,
                                             unsigned lds_off, int tid) {
    uint64_t g = (uint64_t)(uintptr_t)(Wt + (size_t)row * DIM) + (unsigned)(tid * 16);
    unsigned  l = lds_off + (unsigned)(tid * 16);
    asm volatile("global_load_async_to_lds_b128 %0, %1, off"
                 :: "v"(l),          "v"(g)          : "memory");
    asm volatile("global_load_async_to_lds_b128 %0, %1, off"
                 :: "v"(l + 4096u),  "v"(g + 4096u)  : "memory");
}

__device__ __forceinline__ v4f ldrow(const float* __restrict__ W, int r, int q) {
    return ((const v4f*)(W + (size_t)r * DIM))[q];
}

// ---------------------------------------------------------------------------
// Pass 2: each block accumulates SPB samples into a private [DIM] partial.
// Thread owns columns [tid*4, tid*4+4) and [1024+tid*4, ...): fully coalesced.
// ---------------------------------------------------------------------------
__global__ __launch_bounds__(256) void accum_kernel(
    const float* __restrict__ Wx,  const float* __restrict__ Wy,
    const float* __restrict__ Wz,  const float* __restrict__ Wm,
    const float* __restrict__ Wxj, const float* __restrict__ Wyj,
    const float* __restrict__ Wzj, const float* __restrict__ Wmj,
    const float* __restrict__ Wt,  const int* __restrict__ idx,
    float* __restrict__ partials) {

    __shared__ float tbuf[2][DIM];        // 16 KB double buffer for W_t rows
    __shared__ int   sidx[9 * SPB];       // per-block index cache

    const int tid = threadIdx.x;
    const int b   = blockIdx.x;
    const int n0  = b * SPB;

    // cooperative load of this block's 9*SPB indices into LDS
    for (int i = tid; i < 9 * SPB; i += 256) {
        int k = i / SPB, o = i - k * SPB;
        sidx[i] = idx[k * N_SAMP + n0 + o];
    }
    __syncthreads();

    const unsigned tb = (unsigned)(uintptr_t)(&tbuf[0][0]);   // LDS byte offset

    v4f acc0 = (v4f)0.0f, acc1 = (v4f)0.0f;

    prefetch_row(Wt, sidx[8 * SPB + 0], tb, tid);             // ASYNCcnt = 2

    for (int s = 0; s < SPB; ++s) {
        int sn = (s + 1 < SPB) ? s + 1 : s;
        prefetch_row(Wt, sidx[8 * SPB + sn], tb + (unsigned)(((s + 1) & 1) * 8192), tid);

        int jx = sidx[0 * SPB + s], jy = sidx[1 * SPB + s];
        int jz = sidx[2 * SPB + s], jm = sidx[3 * SPB + s];
        int kx = sidx[4 * SPB + s], ky = sidx[5 * SPB + s];
        int kz = sidx[6 * SPB + s], km = sidx[7 * SPB + s];

        v4f x0 = ldrow(Wx, jx, tid),       x1 = ldrow(Wx, jx, tid + 256);
        v4f y0 = ldrow(Wy, jy, tid),       y1 = ldrow(Wy, jy, tid + 256);
        v4f z0 = ldrow(Wz, jz, tid),       z1 = ldrow(Wz, jz, tid + 256);
        v4f m0 = ldrow(Wm, jm, tid),       m1 = ldrow(Wm, jm, tid + 256);
        v4f a0 = ldrow(Wxj, kx, tid),      a1 = ldrow(Wxj, kx, tid + 256);
        v4f b0 = ldrow(Wyj, ky, tid),      b1 = ldrow(Wyj, ky, tid + 256);
        v4f c0 = ldrow(Wzj, kz, tid),      c1 = ldrow(Wzj, kz, tid + 256);
        v4f d0 = ldrow(Wmj, km, tid),      d1 = ldrow(Wmj, km, tid + 256);

        // wait for this sample's W_t row DMA (leaves next row's 2 ops in flight)
        asm volatile("s_wait_asynccnt 0x2" ::: "memory");
        const v4f* tv = (const v4f*)(&tbuf[s & 1][0]);
        v4f t0v = tv[tid], t1v = tv[tid + 256];

        acc0 += ((x0 + y0 + z0) * m0 + (a0 + b0 + c0) * d0) * t0v;
        acc1 += ((x1 + y1 + z1) * m1 + (a1 + b1 + c1) * d1) * t1v;

        // ensure LDS reads of this buffer retired before async engine reuses it
        asm volatile("s_wait_dscnt 0x0" ::: "memory");
    }

    v4f* pr = (v4f*)(partials + (size_t)b * DIM);
    pr[tid]       = acc0;
    pr[tid + 256] = acc1;
}

// ---------------------------------------------------------------------------
// Pass 3: out[d] = tanh( sum_p partials[p][d] ) as ones(16x4) x B(4x16) WMMA.
// A is all-ones so any K-permutation of the B lane layout sums correctly;
// partials are integers < 2^24, so f32 WMMA accumulation is exact.
// One wave per 16-column tile; 128 waves total.
// ---------------------------------------------------------------------------
__global__ __launch_bounds__(128) void reduce_kernel(const float* __restrict__ partials,
                                                     float* __restrict__ out) {
    const int lane = threadIdx.x & 31;
    const int wave = blockIdx.x * (blockDim.x >> 5) + (threadIdx.x >> 5);
    const int d    = wave * 16 + (lane & 15);
    const int kh   = lane >> 4;            // which pair of K-rows this half-wave holds

    v2f A; A[0] = 1.0f; A[1] = 1.0f;       // 16x4 all-ones
    v8f C = {};
    for (int p0 = 0; p0 < P_BLOCKS; p0 += 4) {
        v2f B;
        B[0] = partials[(size_t)(p0 + 2 * kh)     * DIM + d];
        B[1] = partials[(size_t)(p0 + 2 * kh + 1) * DIM + d];
        C = __builtin_amdgcn_wmma_f32_16x16x4_f32(false, A, false, B,
                                                  (short)0, C, false, false);
    }
    if (lane < 16) out[d] = tanhf(C[0]);   // row M=0 of D
}

// ---------------------------------------------------------------------------
extern "C" void kernel_launch(void* const* d_in, const int* in_sizes, int n_in,
                              void* d_out, int out_size, void* d_ws, size_t ws_size,
                              hipStream_t stream) {
    const float* inp = (const float*)d_in[0];
    const float* Wx  = (const float*)d_in[1];
    const float* Wy  = (const float*)d_in[2];
    const float* Wz  = (const float*)d_in[3];
    const float* Wm  = (const float*)d_in[4];
    const float* Wxj = (const float*)d_in[5];
    const float* Wyj = (const float*)d_in[6];
    const float* Wzj = (const float*)d_in[7];
    const float* Wmj = (const float*)d_in[8];
    const float* Wt  = (const float*)d_in[9];

    float* partials = (float*)d_ws;                                    // 2 MB
    int*   idx      = (int*)((char*)d_ws + (size_t)P_BLOCKS * DIM * sizeof(float));

    idx_kernel<<<N_SAMP / 256, 256, 0, stream>>>(inp, idx);
    accum_kernel<<<P_BLOCKS, 256, 0, stream>>>(Wx, Wy, Wz, Wm, Wxj, Wyj, Wzj, Wmj,
                                               Wt, idx, partials);
    reduce_kernel<<<DIM / 16 / 4, 128, 0, stream>>>(partials, (float*)d_out);
}